// CrispComposition_65712999629449
// MI455X (gfx1250) — compile-verified
//
#include <hip/hip_runtime.h>
#include <stdint.h>

#define BB   64
#define IN   2048
#define OUTN 2048
#define TO   128   // out-columns per block
#define KI   64    // i-values per LDS stage
#define NSTG 2     // stages per block -> KC = 128 i-values per block
#define MT_S 68    // padded row stride for m tile (bank-conflict-free)

#if defined(__gfx1250__) && __has_builtin(__builtin_amdgcn_global_load_async_to_lds_b128)
#define ASYNC_LDS 1
#else
#define ASYNC_LDS 0
#endif

typedef int v4i __attribute__((vector_size(16)));
typedef __attribute__((address_space(3))) v4i lds_v4i;

// Copy 16 bytes global -> LDS for this lane (async, tracked by ASYNCcnt).
__device__ __forceinline__ void cp16(float* lds, const float* g) {
#if ASYNC_LDS
    __builtin_amdgcn_global_load_async_to_lds_b128(
        (v4i*)(uintptr_t)g,            // global address (flat pointer per builtin sig)
        (lds_v4i*)(uintptr_t)lds,      // LDS address (low 32 bits of flat == LDS offset)
        /*offset=*/0, /*cpol=*/0);
#else
    *(float4*)lds = *(const float4*)g;
#endif
}

__device__ __forceinline__ void wait_async_then_barrier() {
#if ASYNC_LDS
#if __has_builtin(__builtin_amdgcn_s_wait_asynccnt)
    __builtin_amdgcn_s_wait_asynccnt(0);
#else
    asm volatile("s_wait_asynccnt 0" ::: "memory");
#endif
#endif
    __syncthreads();
}

__global__ void CrispComposition_init_kernel(float* __restrict__ out) {
    int idx = blockIdx.x * blockDim.x + threadIdx.x;
    if (idx < BB * OUTN) out[idx] = 0.0f;   // result >= 0 always (inputs in [0,1))
}

__global__ __launch_bounds__(256)
void CrispComposition_maxmin_kernel(const float* __restrict__ m,
                                    const float* __restrict__ w,
                                    float* __restrict__ out) {
    __shared__ float mt[BB * MT_S];  // m tile, [b][i], padded stride
    __shared__ float wt[KI * TO];    // w tile, [i][o]

    const int tid    = threadIdx.x;
    const int tx     = tid & 15;     // column group: cols 4*tx..+3 and 64+4*tx..+3
    const int ty     = tid >> 4;     // row group: rows 4*ty..4*ty+3
    const int o_base = blockIdx.x * TO;

    float acc[32];
#pragma unroll
    for (int j = 0; j < 32; ++j) acc[j] = 0.0f;

    for (int s = 0; s < NSTG; ++s) {
        const int i_base = (blockIdx.y * NSTG + s) * KI;
        if (s != 0) __syncthreads();   // all readers of previous stage done

        // ---- stage m tile: 64 rows x 64 i (4 float4/thread, coalesced) ----
#pragma unroll
        for (int k = 0; k < 4; ++k) {
            int g  = tid + 256 * k;    // 0..1023 float4 groups
            int b  = g >> 4;           // 0..63
            int i4 = g & 15;           // 0..15
            cp16(&mt[b * MT_S + 4 * i4], &m[b * IN + i_base + 4 * i4]);
        }
        // ---- stage w tile: 64 i x 128 cols (8 float4/thread, coalesced) ----
#pragma unroll
        for (int k = 0; k < 8; ++k) {
            int g  = tid + 256 * k;    // 0..2047 float4 groups
            int i  = g >> 5;           // 0..63
            int o4 = g & 31;           // 0..31
            cp16(&wt[i * TO + 4 * o4], &w[(i_base + i) * OUTN + o_base + 4 * o4]);
        }
        wait_async_then_barrier();

        // ---- compute: 4 rows x 8 cols per thread over KI i-values ----
#pragma unroll 4
        for (int i = 0; i < KI; ++i) {
            const float4 wa = *(const float4*)&wt[i * TO + 4 * tx];
            const float4 wb = *(const float4*)&wt[i * TO + 64 + 4 * tx];
            float wv[8] = {wa.x, wa.y, wa.z, wa.w, wb.x, wb.y, wb.z, wb.w};
            float mv[4];
#pragma unroll
            for (int r = 0; r < 4; ++r)
                mv[r] = mt[(4 * ty + r) * MT_S + i];   // LDS broadcast
#pragma unroll
            for (int r = 0; r < 4; ++r) {
#pragma unroll
                for (int c = 0; c < 8; ++c) {
                    acc[r * 8 + c] = fmaxf(acc[r * 8 + c], fminf(mv[r], wv[c]));
                }
            }
        }
    }

    // ---- combine partials: nonneg-float max == uint bit max (idempotent, replay-safe) ----
    unsigned* outu = (unsigned*)out;
#pragma unroll
    for (int r = 0; r < 4; ++r) {
        int b = 4 * ty + r;
#pragma unroll
        for (int c = 0; c < 8; ++c) {
            int o = o_base + ((c < 4) ? (4 * tx + c) : (64 + 4 * tx + (c - 4)));
            atomicMax(&outu[b * OUTN + o], __float_as_uint(acc[r * 8 + c]));
        }
    }
}

extern "C" void kernel_launch(void* const* d_in, const int* in_sizes, int n_in,
                              void* d_out, int out_size, void* d_ws, size_t ws_size,
                              hipStream_t stream) {
    const float* m = (const float*)d_in[0];   // [64, 2048]
    const float* w = (const float*)d_in[1];   // [2048, 2048]
    float* out = (float*)d_out;               // [64, 2048]

    CrispComposition_init_kernel<<<(BB * OUTN + 255) / 256, 256, 0, stream>>>(out);

    dim3 grid(OUTN / TO, IN / (NSTG * KI));   // (16, 16) = 256 blocks, 2048 waves
    CrispComposition_maxmin_kernel<<<grid, 256, 0, stream>>>(m, w, out);
}